// GPEncoder_27960237097320
// MI455X (gfx1250) — compile-verified
//
#include <hip/hip_runtime.h>
#include <cstdint>
#include <cstddef>

typedef float v2f  __attribute__((ext_vector_type(2)));
typedef float v4f  __attribute__((ext_vector_type(4)));
typedef float v8f  __attribute__((ext_vector_type(8)));
typedef unsigned int u32x4 __attribute__((ext_vector_type(4)));
typedef int   i32x4 __attribute__((ext_vector_type(4)));
typedef int   i32x8 __attribute__((ext_vector_type(8)));

typedef __attribute__((address_space(3))) float as3_float;
typedef __attribute__((address_space(3))) v2f   as3_v2f;

#define NPIX  (512 * 512)   // H*W
#define NB    4             // batch
#define KK    36            // 6*6 inner matrix elements

// ---------------------------------------------------------------------------
// Kernel 1: S (B,H,W,36) -> S_out (B,36,H,W)
// Per 32-lane wave: TDM-load a 16-pixel x 36-float slab into LDS, then use
// v_wmma_f32_16x16x4_f32 with identity-B as an exact f32 tile transposer.
//   A layout (16x4 f32): lane L<16 holds row M=L, K=0..1; lane L+16 holds K=2..3
//   D layout (16x16 f32): lane n holds column n for rows 0..7 (lanes 16..31: rows 8..15)
// => after D = X * I16, lane n holds S[kk=c0+n] for 8 consecutive pixels:
//    two contiguous b128 stores per lane into the output plane.
// The LDS slice pointer is laundered through inline asm: clang cannot see the
// TDM write to `lds` and otherwise folds the reads to zero; the opaque AS(3)
// pointer forces real ds_load_b64s while keeping them below s_wait_tensorcnt.
// ---------------------------------------------------------------------------
__global__ __launch_bounds__(256) void s_transpose_wmma(
    const float* __restrict__ S, float* __restrict__ outS) {
  __shared__ float lds[8 * 16 * KK];          // 8 waves * 16 rows * 36 f32 = 18 KB

  const int lane = threadIdx.x & 31;
  const int wave = threadIdx.x >> 5;
  const long tile = (long)blockIdx.x * 8 + wave;   // one 16-pixel tile per wave
  const long p0g  = tile * 16;                      // global pixel = b*NPIX + p
  const int  b    = (int)(p0g / NPIX);
  const long p0   = p0g - (long)b * NPIX;           // pixel within batch image

  // ---- TDM: DMA 16 rows x 36 f32 (contiguous 2304 B) into this wave's slice
  // `lds` is the only static LDS object -> base offset 0.
  const unsigned ldsOff = (unsigned)(wave * 16 * KK * 4);
  const unsigned long long ga =
      (unsigned long long)(uintptr_t)(S + (size_t)p0g * KK);
  u32x4 g0;
  g0[0] = 1u;                                           // count=1 (valid user D#)
  g0[1] = ldsOff;                                       // lds_addr (bytes)
  g0[2] = (unsigned)(ga & 0xFFFFFFFFull);               // global_addr[31:0]
  g0[3] = (unsigned)((ga >> 32) & 0x1FFFFFFull)         // global_addr[56:32]
          | (2u << 30);                                 // type = 2 ("image")
  i32x8 g1;
  g1[0] = (2 << 16);            // data_size = 2 -> 4-byte elements; mask/flags 0
  g1[1] = (KK << 16);           // tensor_dim0[15:0]=36 (bits 63:48)
  g1[2] = (16 << 16);           // tensor_dim0 hi=0 | tensor_dim1 lo=16 rows
  g1[3] = (KK << 16);           // tensor_dim1 hi=0 | tile_dim0=36
  g1[4] = 16;                   // tile_dim1=16 rows, tile_dim2=0
  g1[5] = KK;                   // tensor_dim0_stride lo32 = 36 elements
  g1[6] = 0;                    // stride0 hi, stride1 lo
  g1[7] = 0;
  const i32x4 gz4 = {0, 0, 0, 0};                       // 2-D: groups 2/3 unused
  const i32x8 gz8 = {0, 0, 0, 0, 0, 0, 0, 0};           // extra group (clang-23 form)
  __builtin_amdgcn_tensor_load_to_lds(g0, g1, gz4, gz4, gz8, 0);
  __builtin_amdgcn_s_wait_tensorcnt(0);

  const int r  = lane & 15;     // A row (pixel) == B/D column index n
  const int hi = lane >> 4;     // lane half selects K 2..3 / rows 8..15

  // Opaque addrspace(3) pointer to this lane's LDS row (see header comment).
  as3_float* row = (as3_float*)&lds[wave * 16 * KK + r * KK];
  asm volatile("" : "+v"(row));

  #pragma unroll
  for (int c0 = 0; c0 < KK; c0 += 16) {
    v8f acc = {0.f, 0.f, 0.f, 0.f, 0.f, 0.f, 0.f, 0.f};
    const int nk = (KK - c0 >= 16) ? 4 : ((KK - c0 + 3) >> 2);  // 4,4,1
    #pragma unroll
    for (int k = 0; k < 4; ++k) {
      if (k >= nk) break;
      const int c = c0 + 4 * k + 2 * hi;               // even -> 8B-aligned pair
      v2f a = {0.0f, 0.0f};
      if (c < KK) a = *(const as3_v2f*)(row + c);      // ds_load_b64
      v2f bv;                                           // identity rows 4k..4k+3
      bv.x = (r == 4 * k + 2 * hi)     ? 1.0f : 0.0f;
      bv.y = (r == 4 * k + 2 * hi + 1) ? 1.0f : 0.0f;
      acc = __builtin_amdgcn_wmma_f32_16x16x4_f32(
          /*neg_a=*/false, a, /*neg_b=*/false, bv,
          /*c_mod=*/(short)0, acc, /*reuse_a=*/false, /*reuse_b=*/false);
    }
    const int kk = c0 + r;                              // output component plane
    if (kk < KK) {
      float* dst = outS + ((size_t)b * KK + kk) * (size_t)NPIX + p0 + hi * 8;
      v4f lo = {acc[0], acc[1], acc[2], acc[3]};
      v4f hf = {acc[4], acc[5], acc[6], acc[7]};
      *(v4f*)(dst)     = lo;                            // 16B aligned: p0 % 16 == 0
      *(v4f*)(dst + 4) = hf;
    }
  }
}

// ---------------------------------------------------------------------------
// Kernel 2: per-pixel 6x6 algebra. One thread = one pixel (32 pixels/wave).
//   LU(D) (no pivot; D ~ 2I + 0.1N -> diagonally dominant)
//   X = D^{-1} R[:3,:]^T  (3 solves)      S1L = R @ X  (left 6x3 block of S_1)
//   Q = Syx @ Sxx^{-1}    y' = my + Q(x-mx)
//   z = [x-mx, Q(x-mx)];  t = R z;  t[:3] *= -1;  out = R^T t + m
// ---------------------------------------------------------------------------
__global__ __launch_bounds__(256) void pixel_math(
    const float* __restrict__ xin, const float* __restrict__ m,
    const float* __restrict__ R,   const float* __restrict__ Dm,
    float* __restrict__ out_x, float* __restrict__ out_y) {
  const long q = (long)blockIdx.x * blockDim.x + threadIdx.x;  // global pixel
  if (q >= (long)NB * NPIX) return;
  const int  b = (int)(q / NPIX);
  const long p = q - (long)b * NPIX;

  float Dv[36], Rv[36];
  {
    const v4f* Dp = (const v4f*)(Dm + (size_t)q * KK);   // 144 B rows -> 16B aligned
    const v4f* Rp = (const v4f*)(R  + (size_t)q * KK);
    #pragma unroll
    for (int i = 0; i < 9; ++i) {
      v4f d4 = Dp[i], r4 = Rp[i];
      Dv[4*i+0] = d4.x; Dv[4*i+1] = d4.y; Dv[4*i+2] = d4.z; Dv[4*i+3] = d4.w;
      Rv[4*i+0] = r4.x; Rv[4*i+1] = r4.y; Rv[4*i+2] = r4.z; Rv[4*i+3] = r4.w;
    }
  }

  // In-place LU factorization of D (no pivoting)
  #pragma unroll
  for (int k = 0; k < 6; ++k) {
    const float piv = 1.0f / Dv[k*6+k];
    #pragma unroll
    for (int i = k + 1; i < 6; ++i) {
      const float f = Dv[i*6+k] * piv;
      Dv[i*6+k] = f;
      #pragma unroll
      for (int j = k + 1; j < 6; ++j) Dv[i*6+j] -= f * Dv[k*6+j];
    }
  }

  // X = D^{-1} * (R[:3,:])^T : column j of RHS is row j of R
  float X[6][3];
  #pragma unroll
  for (int j = 0; j < 3; ++j) {
    float v[6];
    #pragma unroll
    for (int t = 0; t < 6; ++t) v[t] = Rv[j*6 + t];
    #pragma unroll
    for (int i = 1; i < 6; ++i) {                 // forward (unit lower)
      float s = v[i];
      #pragma unroll
      for (int t = 0; t < 6; ++t) if (t < i) s -= Dv[i*6+t] * v[t];
      v[i] = s;
    }
    #pragma unroll
    for (int i = 5; i >= 0; --i) {                // backward (upper)
      float s = v[i];
      #pragma unroll
      for (int t = 0; t < 6; ++t) if (t > i) s -= Dv[i*6+t] * v[t];
      v[i] = s / Dv[i*6+i];
    }
    #pragma unroll
    for (int t = 0; t < 6; ++t) X[t][j] = v[t];
  }

  // S1L = R @ X : left 6x3 block of S_1 = R D^{-1} R^T
  float S1L[6][3];
  #pragma unroll
  for (int i = 0; i < 6; ++i)
    #pragma unroll
    for (int j = 0; j < 3; ++j) {
      float s = 0.f;
      #pragma unroll
      for (int t = 0; t < 6; ++t) s += Rv[i*6+t] * X[t][j];
      S1L[i][j] = s;
    }

  // Sxx^{-1} via adjugate
  const float a00 = S1L[0][0], a01 = S1L[0][1], a02 = S1L[0][2];
  const float a10 = S1L[1][0], a11 = S1L[1][1], a12 = S1L[1][2];
  const float a20 = S1L[2][0], a21 = S1L[2][1], a22 = S1L[2][2];
  const float c00 = a11*a22 - a12*a21;
  const float c01 = a12*a20 - a10*a22;
  const float c02 = a10*a21 - a11*a20;
  const float idet = 1.0f / (a00*c00 + a01*c01 + a02*c02);
  float inv[3][3];
  inv[0][0] = c00*idet; inv[0][1] = (a02*a21 - a01*a22)*idet; inv[0][2] = (a01*a12 - a02*a11)*idet;
  inv[1][0] = c01*idet; inv[1][1] = (a00*a22 - a02*a20)*idet; inv[1][2] = (a02*a10 - a00*a12)*idet;
  inv[2][0] = c02*idet; inv[2][1] = (a01*a20 - a00*a21)*idet; inv[2][2] = (a00*a11 - a01*a10)*idet;

  // Q = Syx @ Sxx^{-1}
  float Q[3][3];
  #pragma unroll
  for (int i = 0; i < 3; ++i)
    #pragma unroll
    for (int j = 0; j < 3; ++j)
      Q[i][j] = S1L[3+i][0]*inv[0][j] + S1L[3+i][1]*inv[1][j] + S1L[3+i][2]*inv[2][j];

  // Load m (6 planes) and x (3 planes); coalesced along pixel axis
  float m6[6], dx[3];
  #pragma unroll
  for (int c = 0; c < 6; ++c) m6[c] = m[((size_t)b*6 + c) * NPIX + p];
  #pragma unroll
  for (int c = 0; c < 3; ++c) dx[c] = xin[((size_t)b*3 + c) * NPIX + p] - m6[c];

  float z[6];
  z[0] = dx[0]; z[1] = dx[1]; z[2] = dx[2];
  #pragma unroll
  for (int i = 0; i < 3; ++i)
    z[3+i] = Q[i][0]*dx[0] + Q[i][1]*dx[1] + Q[i][2]*dx[2];

  // t = R z ; flip first 3 ; u = R^T t ; out = u + m
  float t6[6];
  #pragma unroll
  for (int n = 0; n < 6; ++n) {
    float s = 0.f;
    #pragma unroll
    for (int mm = 0; mm < 6; ++mm) s += Rv[n*6+mm] * z[mm];
    t6[n] = (n < 3) ? -s : s;
  }
  #pragma unroll
  for (int n = 0; n < 6; ++n) {
    float s = 0.f;
    #pragma unroll
    for (int mm = 0; mm < 6; ++mm) s += Rv[mm*6+n] * t6[mm];
    const float zo = s + m6[n];
    if (n < 3) out_x[((size_t)b*3 + n)     * NPIX + p] = zo;
    else       out_y[((size_t)b*3 + n - 3) * NPIX + p] = zo;
  }
}

// ---------------------------------------------------------------------------
extern "C" void kernel_launch(void* const* d_in, const int* in_sizes, int n_in,
                              void* d_out, int out_size, void* d_ws, size_t ws_size,
                              hipStream_t stream) {
  (void)in_sizes; (void)n_in; (void)out_size; (void)d_ws; (void)ws_size;
  const float* x = (const float*)d_in[0];
  /* d_in[1] = y : unused by the reference computation */
  const float* m = (const float*)d_in[2];
  const float* S = (const float*)d_in[3];
  const float* R = (const float*)d_in[4];
  const float* D = (const float*)d_in[5];

  float* out = (float*)d_out;
  const size_t BN = (size_t)NB * NPIX;          // 1,048,576 pixels
  float* out_x = out;                            // (B,3,H,W)
  float* out_y = out + 3 * BN;                   // (B,3,H,W)
  float* out_m = out + 6 * BN;                   // (B,6,H,W) passthrough
  float* out_S = out + 12 * BN;                  // (B,36,H,W)

  (void)hipMemcpyAsync(out_m, m, 6 * BN * sizeof(float),
                       hipMemcpyDeviceToDevice, stream);

  pixel_math<<<dim3((unsigned)((BN + 255) / 256)), dim3(256), 0, stream>>>(
      x, m, R, D, out_x, out_y);

  // 8 waves/block, one 16-pixel tile per wave -> BN/128 blocks
  s_transpose_wmma<<<dim3((unsigned)(BN / 128)), dim3(256), 0, stream>>>(
      S, out_S);
}